// FastNNAttentionHead_24249385353815
// MI455X (gfx1250) — compile-verified
//
#include <hip/hip_runtime.h>
#include <hip/hip_bf16.h>
#include <cmath>

// FastNNAttentionHead for MI455X (gfx1250, wave32).
// Pipeline:
//  K1 (x3): WMMA f32 16x16x4 GEMM  Out[m,n] = sum_k X[m,k] * W[n,k]   (A, B, V)
//  K2     : async-LDS-tiled pairwise GELU reduction -> S[b,i,j] (causal tiles)
//           (global_load_async_to_lds_b128 + s_wait_asynccnt staging)
//  K3     : causal softmax over j (block reduction), wei overwrites S
//  K4     : WMMA f32 16x16x4 GEMM  out = wei @ V (K-loop truncated past diagonal)

typedef __attribute__((ext_vector_type(2))) float v2f;
typedef __attribute__((ext_vector_type(8))) float v8f;

static constexpr int Bb = 2;     // batch
static constexpr int T  = 512;   // sequence
static constexpr int C  = 256;   // channels
static constexpr int HS = 64;    // head size
static constexpr int BT = Bb * T;
static constexpr int CP = C + 4; // LDS row stride: 1040B (16B aligned, bank-safe)

// ---------------------------------------------------------------------------
// Kernel 1: generic Out[m,n] = sum_k X[m,k] * W[n*wstride + wcoff + k]
// One wave (32 threads) computes one 16x16 output tile via v_wmma_f32_16x16x4_f32.
// A-frag (16x4): lane L -> (M = L&15, K = k0 + 2*(L>>4) + v), contiguous pair.
// B-frag (4x16): lane L -> (N = L&15, K = k0 + 2*(L>>4) + v), contiguous pair.
// D (16x16):     vgpr r -> (M = r + 8*(L>>4), N = L&15).
// ---------------------------------------------------------------------------
__global__ __launch_bounds__(32)
void gemm_xwT_wmma(const float* __restrict__ X, const float* __restrict__ W,
                   float* __restrict__ Out, int N, int K, int wstride, int wcoff) {
  const int lane = threadIdx.x;
  const int lo = lane & 15;
  const int hi = lane >> 4;
  const int m = blockIdx.y * 16 + lo;
  const int n = blockIdx.x * 16 + lo;

  const float* __restrict__ xrow = X + (size_t)m * K;
  const float* __restrict__ wrow = W + (size_t)n * wstride + wcoff;

  v8f acc = {};
  for (int k0 = 0; k0 < K; k0 += 4) {
    const int ka = k0 + 2 * hi;
    const float2 av = *(const float2*)(xrow + ka);
    const float2 bv = *(const float2*)(wrow + ka);
    v2f a;  a.x = av.x;  a.y = av.y;
    v2f bf; bf.x = bv.x; bf.y = bv.y;
    acc = __builtin_amdgcn_wmma_f32_16x16x4_f32(
        /*neg_a=*/false, a, /*neg_b=*/false, bf,
        /*c_mod=*/(short)0, acc, /*reuse_a=*/false, /*reuse_b=*/false);
  }

  const int orow0 = blockIdx.y * 16 + 8 * hi;
  const int ocol  = blockIdx.x * 16 + lo;
#pragma unroll
  for (int r = 0; r < 8; ++r) {
    Out[(size_t)(orow0 + r) * N + ocol] = acc[r];
  }
}

// ---------------------------------------------------------------------------
// Kernel 2: S[b,i,j] = (sum_h w2[h]*gelu(A[b,i,h]+B[b,j,h]) + b2) * c^-0.5
// Block = 256 threads = one 16x16 (i,j) tile. Tiles fully above the diagonal
// are skipped; softmax masks by index so their S storage is never read.
// A/B tiles (16x256 f32 each) staged global->LDS with ASYNC copies (no VGPR
// round trip), tracked by ASYNCcnt.
// ---------------------------------------------------------------------------
__device__ __forceinline__ float gelu_exact(float v) {
  // exact GELU: 0.5*v*(1+erf(v/sqrt(2)))  (matches jax.nn.gelu approximate=False)
  return 0.5f * v * (1.0f + erff(v * 0.70710678118654752f));
}

__global__ __launch_bounds__(256)
void scores_kernel(const float* __restrict__ Afeat, const float* __restrict__ Bfeat,
                   const float* __restrict__ w2, const float* __restrict__ b2,
                   float* __restrict__ S) {
  const int jt = blockIdx.x;
  const int it = blockIdx.y;
  const int bb = blockIdx.z;
  if (jt > it) return;  // whole tile strictly masked

  __shared__ float sA[16][CP];  // stride 260 floats: 16B-aligned rows, no bank conflicts
  __shared__ float sB[16][CP];
  __shared__ float sw[C];

  const int tid = threadIdx.x;

  // Async global->LDS staging: 16 rows x 64 chunks of 16B per tile array.
  // Generic pointer low 32 bits == LDS byte offset (LDS aperture, ISA 10.2).
  for (int idx = tid; idx < 16 * 64; idx += 256) {
    const int row = idx >> 6;
    const int c4  = (idx & 63) << 2;  // float index, 16B aligned
    const float* gA = Afeat + (size_t)(bb * T + it * 16 + row) * C + c4;
    const float* gB = Bfeat + (size_t)(bb * T + jt * 16 + row) * C + c4;
    const unsigned ldsA = (unsigned)(size_t)&sA[row][c4];
    const unsigned ldsB = (unsigned)(size_t)&sB[row][c4];
    asm volatile("global_load_async_to_lds_b128 %0, %1, off"
                 :: "v"(ldsA), "v"(gA) : "memory");
    asm volatile("global_load_async_to_lds_b128 %0, %1, off"
                 :: "v"(ldsB), "v"(gB) : "memory");
  }
  sw[tid] = w2[tid];
  asm volatile("s_wait_asynccnt 0x0" ::: "memory");
  __syncthreads();

  const int ti = tid & 15;   // i within tile
  const int tj = tid >> 4;   // j within tile
  float acc = 0.0f;
#pragma unroll 4
  for (int h = 0; h < C; ++h) {
    const float v = sA[ti][h] + sB[tj][h];
    acc = fmaf(sw[h], gelu_exact(v), acc);
  }
  const float sv = (acc + b2[0]) * 0.0625f;  // c^-0.5 = 1/16
  S[(size_t)bb * T * T + (size_t)(it * 16 + ti) * T + (jt * 16 + tj)] = sv;
}

// ---------------------------------------------------------------------------
// Kernel 3: causal softmax over j (strict j < i), in-place: S -> wei.
// One block (256 threads) per row; each thread owns j and j+256.
// Row i==0 has no valid entries -> all zeros (matches mask-multiply in ref).
// ---------------------------------------------------------------------------
__global__ __launch_bounds__(256)
void softmax_kernel(float* __restrict__ S) {
  const int flat = blockIdx.x;
  const int bb = flat >> 9;         // /512
  const int i  = flat & (T - 1);
  float* __restrict__ row = S + (size_t)bb * T * T + (size_t)i * T;

  const int tid = threadIdx.x;
  const int j0 = tid, j1 = tid + 256;
  const bool v0 = (j0 < i), v1 = (j1 < i);
  const float NEG = -3.4e38f;
  const float s0 = v0 ? row[j0] : NEG;
  const float s1 = v1 ? row[j1] : NEG;

  __shared__ float red[256];
  red[tid] = fmaxf(s0, s1);
  __syncthreads();
  for (int off = 128; off > 0; off >>= 1) {
    if (tid < off) red[tid] = fmaxf(red[tid], red[tid + off]);
    __syncthreads();
  }
  const float mx = red[0];
  __syncthreads();

  const float e0 = v0 ? expf(s0 - mx) : 0.0f;
  const float e1 = v1 ? expf(s1 - mx) : 0.0f;
  red[tid] = e0 + e1;
  __syncthreads();
  for (int off = 128; off > 0; off >>= 1) {
    if (tid < off) red[tid] += red[tid + off];
    __syncthreads();
  }
  const float sum = red[0];
  const float inv = (sum > 0.0f) ? (1.0f / sum) : 0.0f;
  row[j0] = e0 * inv;
  row[j1] = e1 * inv;
}

// ---------------------------------------------------------------------------
// Kernel 4: out[b,i,h] = sum_j wei[b,i,j] * V[b,j,h] via WMMA f32 16x16x4.
// K-loop stops after the diagonal tile (wei is exactly 0 beyond it).
// ---------------------------------------------------------------------------
__global__ __launch_bounds__(32)
void av_wmma(const float* __restrict__ Wei, const float* __restrict__ V,
             float* __restrict__ Out) {
  const int ht = blockIdx.x;   // hs tile (0..3)
  const int it = blockIdx.y;   // i tile  (0..31)
  const int bb = blockIdx.z;
  const int lane = threadIdx.x;
  const int lo = lane & 15;
  const int hi = lane >> 4;

  const float* __restrict__ wrow  = Wei + (size_t)bb * T * T + (size_t)(it * 16 + lo) * T;
  const float* __restrict__ vbase = V + (size_t)bb * T * HS + ht * 16 + lo;

  v8f acc = {};
  const int kmax = it * 16 + 16;  // wei[i, j>=kmax] == 0 for all i in this tile
  for (int k0 = 0; k0 < kmax; k0 += 4) {
    const int ka = k0 + 2 * hi;
    const float2 av = *(const float2*)(wrow + ka);
    v2f a;  a.x = av.x;  a.y = av.y;
    v2f bf; bf.x = vbase[(size_t)ka * HS];
            bf.y = vbase[(size_t)(ka + 1) * HS];
    acc = __builtin_amdgcn_wmma_f32_16x16x4_f32(
        false, a, false, bf, (short)0, acc, false, false);
  }

  const int orow0 = bb * T + it * 16 + 8 * hi;
  const int ocol  = ht * 16 + lo;
#pragma unroll
  for (int r = 0; r < 8; ++r) {
    Out[(size_t)(orow0 + r) * HS + ocol] = acc[r];
  }
}

// ---------------------------------------------------------------------------
extern "C" void kernel_launch(void* const* d_in, const int* in_sizes, int n_in,
                              void* d_out, int out_size, void* d_ws, size_t ws_size,
                              hipStream_t stream) {
  (void)in_sizes; (void)n_in; (void)out_size; (void)ws_size;

  const float* x    = (const float*)d_in[0];  // (2,512,256)
  // d_in[1] = pos_emb, unused by reference
  const float* W1   = (const float*)d_in[2];  // (256,512) row-major [h, 2c]
  const float* w2   = (const float*)d_in[3];  // (256,)
  const float* b2   = (const float*)d_in[4];  // (1,)
  const float* Wv   = (const float*)d_in[5];  // (64,256) row-major [h, c]
  float* out = (float*)d_out;                 // (2,512,64)

  // workspace carve-up (floats): A | B | V | S   (~4.25 MB total)
  float* wsf = (float*)d_ws;
  float* Aw = wsf;                               // 1024*256
  float* Bw = Aw + (size_t)BT * C;               // 1024*256
  float* Vw = Bw + (size_t)BT * C;               // 1024*64
  float* Sw = Vw + (size_t)BT * HS;              // 2*512*512

  // K1: A = x @ W1[:, :c]^T ; B = x @ W1[:, c:]^T ; V = x @ Wv^T
  {
    dim3 gA(C / 16, BT / 16);        // (16, 64)
    gemm_xwT_wmma<<<gA, 32, 0, stream>>>(x, W1, Aw, C, C, 2 * C, 0);
    gemm_xwT_wmma<<<gA, 32, 0, stream>>>(x, W1, Bw, C, C, 2 * C, C);
    dim3 gV(HS / 16, BT / 16);       // (4, 64)
    gemm_xwT_wmma<<<gV, 32, 0, stream>>>(x, Wv, Vw, HS, C, C, 0);
  }

  // K2: pairwise GELU scores (causal tiles only), async-LDS staging
  {
    dim3 g(T / 16, T / 16, Bb);      // (32, 32, 2)
    scores_kernel<<<g, 256, 0, stream>>>(Aw, Bw, w2, b2, Sw);
  }

  // K3: causal softmax, in place
  softmax_kernel<<<Bb * T, 256, 0, stream>>>(Sw);

  // K4: out = wei @ V
  {
    dim3 g(HS / 16, T / 16, Bb);     // (4, 32, 2)
    av_wmma<<<g, 32, 0, stream>>>(Sw, Vw, out);
  }
}